// ResidualBlock_80616536146732
// MI455X (gfx1250) — compile-verified
//
#include <hip/hip_runtime.h>

typedef __attribute__((ext_vector_type(2))) float v2f;
typedef __attribute__((ext_vector_type(4))) float v4f;
typedef __attribute__((ext_vector_type(8))) float v8f;

#define C_CH 256
#define L_LEN 262144
#define KTOT 768          // C_CH * 3  (K-index = ci*3 + k, contiguous in conv_w)
#define NB 128            // output columns per block in pass 1
#define LDS_STRIDE 136    // floats per LDS row: 3-float lead-in + 130 cols + pad
                          // row stride = 544 B (16B aligned); stored col s lives
                          // at float index row*136 + 3 + s, s = g - n0 + 1
#define EPS 1e-5f

// ---------------------------------------------------------------------------
// Pass 1: conv1d as GEMM via V_WMMA_F32_16X16X4_F32.
// Block: 512 threads = 16 waves. Wave (wc,wn): wc in [0,8) -> 32 output rows,
// wn in [0,2) -> 64 output cols. Register tile: 2 (M) x 4 (N) accumulators.
// x tile staged with CDNA5 async global->LDS (ASYNCcnt) path.
// Also emits per-channel partial sum / sumsq for the normalization.
// ---------------------------------------------------------------------------
__global__ __launch_bounds__(512) void conv_wmma_kernel(
    const float* __restrict__ x, const float* __restrict__ w,
    float* __restrict__ out, float* __restrict__ partial)
{
  extern __shared__ float lds_x[];     // [256][LDS_STRIDE] fp32
  const int tid = threadIdx.x;
  const int n0  = blockIdx.x * NB;

  // LDS byte offset of lds_x: low 32 bits of the generic pointer are the
  // wave-relative LDS address (aperture form {SHARED_BASE, lds_offset}).
  const unsigned lds_base = (unsigned)(unsigned long long)lds_x;

  // ---- Async stage: interior columns n0..n0+127, 32 x b128 per row --------
  for (int i = tid; i < C_CH * 32; i += 512) {
    int row = i >> 5;
    int q   = i & 31;                      // 4-float chunk within row
    unsigned loff = lds_base + (unsigned)((row * LDS_STRIDE + 4 + 4 * q) * 4);
    unsigned goff = (unsigned)((row * L_LEN + n0 + 4 * q) * 4);
    asm volatile("global_load_async_to_lds_b128 %0, %1, %2"
                 :: "v"(loff), "v"(goff), "s"(x) : "memory");
  }
  // ---- Async stage: circular halo columns (n0-1) and (n0+128) -------------
  {
    int row  = tid >> 1;
    int side = tid & 1;                    // 0 -> left halo, 1 -> right halo
    int g    = side ? (n0 + NB) : (n0 - 1);
    g = (g < 0) ? (g + L_LEN) : ((g >= L_LEN) ? (g - L_LEN) : g);
    int s = side ? 129 : 0;                // stored column
    unsigned loff = lds_base + (unsigned)((row * LDS_STRIDE + 3 + s) * 4);
    unsigned goff = (unsigned)((row * L_LEN + g) * 4);
    asm volatile("global_load_async_to_lds_b32 %0, %1, %2"
                 :: "v"(loff), "v"(goff), "s"(x) : "memory");
  }
  asm volatile("s_wait_asynccnt 0x0" ::: "memory");
  __syncthreads();

  const int wave = tid >> 5;
  const int lane = tid & 31;
  const int wc   = wave >> 1;          // 0..7  -> 32 channels
  const int wn   = wave & 1;           // 0..1  -> 64 columns
  const int l16  = lane & 15;
  const int hi   = lane >> 4;          // lane half (selects K pair)
  const int co_base   = wc * 32;
  const int nloc_base = wn * 64;

  v8f acc[2][4];
#pragma unroll
  for (int mt = 0; mt < 2; ++mt)
#pragma unroll
    for (int nt = 0; nt < 4; ++nt)
      acc[mt][nt] = (v8f){0.f,0.f,0.f,0.f,0.f,0.f,0.f,0.f};

  // A-fragment base pointers (W rows are 768 contiguous floats).
  const float* wrow0 = w + (size_t)(co_base + l16) * KTOT;        // mt = 0
  const float* wrow1 = w + (size_t)(co_base + 16 + l16) * KTOT;   // mt = 1
  const int koff = hi * 2;   // lanes 0-15 hold K+0/K+1, lanes 16-31 hold K+2/K+3

  for (int kk = 0; kk < KTOT; kk += 4) {
    // Uniform (ci,k) decode for the 4 K-indices of this step (SALU).
    int c0 = (kk + 0) / 3, k0 = (kk + 0) - c0 * 3;
    int c1 = (kk + 1) / 3, k1 = (kk + 1) - c1 * 3;
    int c2 = (kk + 2) / 3, k2 = (kk + 2) - c2 * 3;
    int c3 = (kk + 3) / 3, k3 = (kk + 3) - c3 * 3;
    // Per-lane-half selection (B layout mirrors A: lanes 0-15 -> K0/K1).
    // Element x[ci][n+k-1] lives at float index ci*136 + 3 + (n_local + k).
    int baseA = (hi ? c2 : c0) * LDS_STRIDE + 3 + (hi ? k2 : k0) + nloc_base + l16;
    int baseB = (hi ? c3 : c1) * LDS_STRIDE + 3 + (hi ? k3 : k1) + nloc_base + l16;

    v2f a0 = *(const v2f*)(wrow0 + kk + koff);
    v2f a1 = *(const v2f*)(wrow1 + kk + koff);

#pragma unroll
    for (int nt = 0; nt < 4; ++nt) {
      v2f b;
      b.x = lds_x[baseA + nt * 16];
      b.y = lds_x[baseB + nt * 16];
      acc[0][nt] = __builtin_amdgcn_wmma_f32_16x16x4_f32(
          false, a0, false, b, (short)0, acc[0][nt], false, false);
      acc[1][nt] = __builtin_amdgcn_wmma_f32_16x16x4_f32(
          false, a1, false, b, (short)0, acc[1][nt], false, false);
    }
  }

  // Write conv output + per-channel partial stats (fixed-order -> deterministic).
  const int nbase_out = n0 + nloc_base;
#pragma unroll
  for (int mt = 0; mt < 2; ++mt) {
#pragma unroll
    for (int v = 0; v < 8; ++v) {
      // C/D layout: VGPR v, lanes 0-15 -> row v, lanes 16-31 -> row v+8.
      int ch = co_base + mt * 16 + v + hi * 8;
      float s = 0.f, s2 = 0.f;
#pragma unroll
      for (int nt = 0; nt < 4; ++nt) {
        float val = acc[mt][nt][v];
        out[(size_t)ch * L_LEN + nbase_out + nt * 16 + l16] = val;
        s  += val;
        s2 += val * val;
      }
#pragma unroll
      for (int m = 1; m < 16; m <<= 1) {      // reduce within each 16-lane half
        s  += __shfl_xor(s,  m, 32);
        s2 += __shfl_xor(s2, m, 32);
      }
      if (l16 == 0) {
        int row = blockIdx.x * 2 + wn;        // unique slot -> no atomics
        partial[(size_t)row * 512 + ch]       = s;
        partial[(size_t)row * 512 + 256 + ch] = s2;
      }
    }
  }
}

// ---------------------------------------------------------------------------
// Pass 2: reduce per-block partials -> per-channel mean and rstd.
// One block per channel; fixed-order tree reduction.
// ---------------------------------------------------------------------------
__global__ __launch_bounds__(256) void stats_kernel(
    const float* __restrict__ partial, float* __restrict__ meanv,
    float* __restrict__ rstdv, int rows)
{
  __shared__ float ls[256];
  __shared__ float ls2[256];
  const int c = blockIdx.x;
  const int t = threadIdx.x;
  float s = 0.f, s2 = 0.f;
  for (int r = t; r < rows; r += 256) {
    s  += partial[(size_t)r * 512 + c];
    s2 += partial[(size_t)r * 512 + 256 + c];
  }
  ls[t] = s; ls2[t] = s2;
  __syncthreads();
  for (int off = 128; off > 0; off >>= 1) {
    if (t < off) { ls[t] += ls[t + off]; ls2[t] += ls2[t + off]; }
    __syncthreads();
  }
  if (t == 0) {
    float sum = ls[0], sumsq = ls2[0];
    float mean = sum / (float)L_LEN;
    float var  = (sumsq - sum * mean) / (float)(L_LEN - 1);   // unbiased (ddof=1)
    meanv[c] = mean;
    rstdv[c] = rsqrtf(var + EPS);
  }
}

// ---------------------------------------------------------------------------
// Pass 3: io = (io - mean)*rstd + beta + x, vectorized float4, in place.
// ---------------------------------------------------------------------------
__global__ __launch_bounds__(256) void finalize_kernel(
    const float* __restrict__ x, const float* __restrict__ beta,
    const float* __restrict__ meanv, const float* __restrict__ rstdv,
    float* io)
{
  const long long i = (long long)blockIdx.x * 256 + threadIdx.x; // float4 index
  const int c = (int)(i >> 16);              // L/4 = 65536 float4 per channel
  v4f cv = ((const v4f*)io)[i];
  v4f xv = ((const v4f*)x)[i];
  const float m = meanv[c], r = rstdv[c], b = beta[c];
  v4f o;
  o.x = (cv.x - m) * r + b + xv.x;
  o.y = (cv.y - m) * r + b + xv.y;
  o.z = (cv.z - m) * r + b + xv.z;
  o.w = (cv.w - m) * r + b + xv.w;
  ((v4f*)io)[i] = o;
}

extern "C" void kernel_launch(void* const* d_in, const int* in_sizes, int n_in,
                              void* d_out, int out_size, void* d_ws, size_t ws_size,
                              hipStream_t stream) {
  const float* x    = (const float*)d_in[0];
  const float* w    = (const float*)d_in[1];
  // d_in[2] = gamma: unused in BN mode (reference only adds beta).
  const float* beta = (const float*)d_in[3];
  float* out = (float*)d_out;
  float* ws  = (float*)d_ws;

  float* meanv   = ws;         // [256]
  float* rstdv   = ws + 256;   // [256]
  float* partial = ws + 512;   // [2*(L/NB)][512]  (~8.4 MB)

  const int nblocks1 = L_LEN / NB;                       // 2048
  const size_t ldsBytes = (size_t)C_CH * LDS_STRIDE * sizeof(float); // 139264 B

  conv_wmma_kernel<<<nblocks1, 512, ldsBytes, stream>>>(x, w, out, partial);
  stats_kernel<<<C_CH, 256, 0, stream>>>(partial, meanv, rstdv, nblocks1 * 2);
  finalize_kernel<<<(C_CH * (L_LEN / 4)) / 256, 256, 0, stream>>>(
      x, beta, meanv, rstdv, out);
}